// LaplacianReg_42838003810614
// MI455X (gfx1250) — compile-verified
//
#include <hip/hip_runtime.h>

// Problem constants (match the reference).
#define BATCH  32
#define VTOT   65536
#define KNB    8
#define CCH    3
#define VCHUNK 32   // v's per block

// Async global->LDS copy of 16 bytes per active lane (gfx1250 ASYNC path).
__device__ __forceinline__ void async_copy_b128(unsigned lds_off, const void* gaddr) {
    asm volatile("global_load_async_to_lds_b128 %0, %1, off"
                 :: "v"(lds_off), "v"(gaddr)
                 : "memory");
}

__device__ __forceinline__ void wait_asynccnt0() {
    asm volatile("s_wait_asynccnt 0" ::: "memory");
}

__global__ __launch_bounds__(256)
void LaplacianReg_kernel(const float* __restrict__ out,
                         const float* __restrict__ tgt,
                         const int*   __restrict__ nidx,   // (V, K)
                         const float* __restrict__ nw,     // (V, K)
                         float*       __restrict__ res)    // (B, V, C)
{
    __shared__ __align__(16) int   s_idx[VCHUNK][KNB];   // 1 KB
    __shared__ __align__(16) float s_w  [VCHUNK][KNB];   // 1 KB

    const int tid = threadIdx.x;
    const int v0  = blockIdx.x * VCHUNK;

    // ---- Stage idx/w rows for this v-chunk via async tensor-path copies ----
    // idx and w are (V, K) row-major -> this chunk's rows are 1 KB contiguous each.
    if (tid < 128) {
        if (tid < 64) {
            unsigned lds_off = (unsigned)(unsigned long long)(&s_idx[0][0]) + tid * 16u;
            const char* src  = (const char*)nidx + (size_t)v0 * (KNB * 4) + tid * 16;
            async_copy_b128(lds_off, src);
        } else {
            const int t = tid - 64;
            unsigned lds_off = (unsigned)(unsigned long long)(&s_w[0][0]) + t * 16u;
            const char* src  = (const char*)nw + (size_t)v0 * (KNB * 4) + t * 16;
            async_copy_b128(lds_off, src);
        }
    }
    wait_asynccnt0();
    __syncthreads();

    // ---- Compute ----
    const int vl = tid & 31;           // lane == local v -> coalesced streams
    const int v  = v0 + vl;

    // Per-v neighbor data, loop-invariant across batches: two B128 LDS reads each.
    const int4   i0 = *reinterpret_cast<const int4*  >(&s_idx[vl][0]);
    const int4   i1 = *reinterpret_cast<const int4*  >(&s_idx[vl][4]);
    const float4 w0 = *reinterpret_cast<const float4*>(&s_w[vl][0]);
    const float4 w1 = *reinterpret_cast<const float4*>(&s_w[vl][4]);

    const int   j [KNB] = { i0.x, i0.y, i0.z, i0.w, i1.x, i1.y, i1.z, i1.w };
    const float wv[KNB] = { w0.x, w0.y, w0.z, w0.w, w1.x, w1.y, w1.z, w1.w };

    const int vo = v * CCH;

    #pragma unroll 2
    for (int bi = 0; bi < 4; ++bi) {
        const int b    = (tid >> 5) + bi * 8;          // 8 batches per pass x 4
        const int base = b * (VTOT * CCH);             // fits in 32-bit (max ~6.3M)
        const float* __restrict__ po = out + base;
        const float* __restrict__ pt = tgt + base;

        // lap(out - tgt) accumulated directly (linearity of the Laplacian).
        float a0 = po[vo + 0] - pt[vo + 0];
        float a1 = po[vo + 1] - pt[vo + 1];
        float a2 = po[vo + 2] - pt[vo + 2];

        #pragma unroll
        for (int k = 0; k < KNB; ++k) {
            const int go = j[k] * CCH;                 // random 12B gather, L2-resident
            const float wk = wv[k];
            a0 += wk * (po[go + 0] - pt[go + 0]);
            a1 += wk * (po[go + 1] - pt[go + 1]);
            a2 += wk * (po[go + 2] - pt[go + 2]);
        }

        float* __restrict__ pr = res + base + vo;
        pr[0] = a0 * a0;
        pr[1] = a1 * a1;
        pr[2] = a2 * a2;
    }
}

extern "C" void kernel_launch(void* const* d_in, const int* in_sizes, int n_in,
                              void* d_out, int out_size, void* d_ws, size_t ws_size,
                              hipStream_t stream) {
    (void)in_sizes; (void)n_in; (void)out_size; (void)d_ws; (void)ws_size;
    const float* out_p = (const float*)d_in[0];
    const float* tgt_p = (const float*)d_in[1];
    const int*   idx_p = (const int*)  d_in[2];
    const float* w_p   = (const float*)d_in[3];
    float*       res_p = (float*)d_out;

    const int blocks = VTOT / VCHUNK;   // 2048
    LaplacianReg_kernel<<<blocks, 256, 0, stream>>>(out_p, tgt_p, idx_p, w_p, res_p);
}